// AP_30270929502484
// MI455X (gfx1250) — compile-verified
//
#include <hip/hip_runtime.h>

#define F 128

typedef float v2f __attribute__((ext_vector_type(2)));
typedef float v8f __attribute__((ext_vector_type(8)));

// ---------------------------------------------------------------------------
// Dense GEMM  H[n,128] = X[n,128] @ W[128,128]  via V_WMMA_F32_16X16X4_F32.
// Block = (32,8): 8 waves, wave y owns N-tile y (8 tiles * 16 = 128 cols).
// Each wave: 16x16 output tile, K=128 in steps of 4 -> 32 WMMA ops.
// ---------------------------------------------------------------------------
__global__ __launch_bounds__(256) void gat_gemm_wmma(
    const float* __restrict__ X, const float* __restrict__ W,
    float* __restrict__ H, int nrows)
{
  const int lane  = threadIdx.x;       // 0..31
  const int ntile = threadIdx.y;       // 0..7
  const int l     = lane & 15;
  const int half  = lane >> 4;
  const int mbase = blockIdx.x * 16;

  int row = mbase + l;
  if (row >= nrows) row = nrows - 1;   // clamp reads; stores guarded below
  const int col = ntile * 16 + l;

  const float* xrow = X + (size_t)row * F;
  v8f acc = {};
#pragma unroll
  for (int k = 0; k < F; k += 4) {
    // A 16x4 f32 layout: VGPR0 = {K0 | K2}, VGPR1 = {K1 | K3} across halves
    v2f a, b;
    a.x = xrow[k + 2 * half + 0];
    a.y = xrow[k + 2 * half + 1];
    // B 4x16 f32: row K striped across lanes within a VGPR (symmetric layout)
    b.x = W[(size_t)(k + 2 * half + 0) * F + col];
    b.y = W[(size_t)(k + 2 * half + 1) * F + col];
    acc = __builtin_amdgcn_wmma_f32_16x16x4_f32(
        /*neg_a=*/false, a, /*neg_b=*/false, b,
        /*c_mod=*/(short)0, acc, /*reuse_a=*/false, /*reuse_b=*/false);
  }
  // C/D layout: VGPR r holds M = r (lanes 0-15) / M = 8+r (lanes 16-31)
#pragma unroll
  for (int r = 0; r < 8; ++r) {
    int orow = mbase + r + 8 * half;
    if (orow < nrows) H[(size_t)orow * F + col] = acc[r];
  }
}

// ---------------------------------------------------------------------------
// Hardware f32 atomic add (no return -> STOREcnt; drained by s_endpgm).
// Forces GLOBAL_ATOMIC_ADD_F32 at device scope instead of a CAS loop.
// ---------------------------------------------------------------------------
__device__ __forceinline__ void gfadd(float* p, float v) {
  asm volatile("global_atomic_add_f32 %0, %1, off scope:SCOPE_DEV"
               :: "v"(p), "v"(v) : "memory");
}

// ---------------------------------------------------------------------------
// Per-node attention logits: asrc[n] = h[n].a_src, adst[n] = h[n].a_dst
// One wave per node, 4 features per lane, shuffle reduction.
// ---------------------------------------------------------------------------
__global__ __launch_bounds__(256) void gat_alphas(
    const float* __restrict__ H,
    const float* __restrict__ avs, const float* __restrict__ avd,
    float* __restrict__ asrc, float* __restrict__ adst, int n)
{
  const int wave = (blockIdx.x * blockDim.x + threadIdx.x) >> 5;
  const int lane = threadIdx.x & 31;
  if (wave >= n) return;
  const float* h = H + (size_t)wave * F;
  float s0 = 0.f, s1 = 0.f;
#pragma unroll
  for (int j = 0; j < 4; ++j) {
    float v = h[lane + 32 * j];
    s0 += v * avs[lane + 32 * j];
    s1 += v * avd[lane + 32 * j];
  }
#pragma unroll
  for (int off = 16; off > 0; off >>= 1) {
    s0 += __shfl_xor(s0, off, 32);
    s1 += __shfl_xor(s1, off, 32);
  }
  if (lane == 0) { asrc[wave] = s0; adst[wave] = s1; }
}

// ---------------------------------------------------------------------------
// Init: O[n,f] = bias[f]; segment-max key = 0 (== -inf); segment-sum = 0
// ---------------------------------------------------------------------------
__global__ void gat_init(float* __restrict__ O, const float* __restrict__ bias,
                         unsigned* __restrict__ m, float* __restrict__ s, int n)
{
  const int i = blockIdx.x * blockDim.x + threadIdx.x;
  if (i < n * F) O[i] = bias[i & (F - 1)];
  if (i < n) { m[i] = 0u; s[i] = 0.f; }
}

// monotonic float<->uint mapping so atomicMax(uint) implements float max
__device__ __forceinline__ unsigned fenc(float f) {
  unsigned b = __float_as_uint(f);
  return (b & 0x80000000u) ? ~b : (b | 0x80000000u);
}
__device__ __forceinline__ float fdec(unsigned u) {
  unsigned b = (u & 0x80000000u) ? (u ^ 0x80000000u) : ~u;
  return __uint_as_float(b);
}
__device__ __forceinline__ float leaky(float e) {
  return e > 0.f ? e : 0.2f * e;
}

// edge id >= E encodes self-loop (s = d = id - E)
__global__ void gat_edge_max(const int* __restrict__ src, const int* __restrict__ dst,
                             int E, int n, const float* __restrict__ asrc,
                             const float* __restrict__ adst, unsigned* __restrict__ m)
{
  const int i = blockIdx.x * blockDim.x + threadIdx.x;
  if (i >= E + n) return;
  const int s = (i < E) ? src[i] : (i - E);
  const int d = (i < E) ? dst[i] : (i - E);
  const float e = leaky(asrc[s] + adst[d]);
  atomicMax(m + d, fenc(e));
}

__global__ void gat_edge_sum(const int* __restrict__ src, const int* __restrict__ dst,
                             int E, int n, const float* __restrict__ asrc,
                             const float* __restrict__ adst,
                             const unsigned* __restrict__ m, float* __restrict__ sum)
{
  const int i = blockIdx.x * blockDim.x + threadIdx.x;
  if (i >= E + n) return;
  const int s = (i < E) ? src[i] : (i - E);
  const int d = (i < E) ? dst[i] : (i - E);
  const float e = leaky(asrc[s] + adst[d]);
  gfadd((float*)sum + d, __expf(e - fdec(m[d])));
}

// One wave per edge: lane 0 computes the softmax weight and broadcasts it;
// each lane gathers a float4 of h[src] (coalesced b128) and scatters with
// hardware f32 atomics into O[dst].
__global__ __launch_bounds__(256) void gat_edge_agg(
    const int* __restrict__ src, const int* __restrict__ dst, int E, int n,
    const float* __restrict__ asrc, const float* __restrict__ adst,
    const unsigned* __restrict__ m, const float* __restrict__ sum,
    const float* __restrict__ H, float* __restrict__ O)
{
  const int gid  = blockIdx.x * blockDim.x + threadIdx.x;
  const int edge = gid >> 5;              // uniform across the wave
  const int lane = threadIdx.x & 31;
  if (edge >= E + n) return;              // whole wave exits together

  int s = 0, d = 0;
  float w = 0.f;
  if (lane == 0) {
    s = (edge < E) ? src[edge] : (edge - E);
    d = (edge < E) ? dst[edge] : (edge - E);
    const float e = leaky(asrc[s] + adst[d]);
    w = __expf(e - fdec(m[d])) / sum[d];
  }
  s = __shfl(s, 0, 32);
  d = __shfl(d, 0, 32);
  w = __shfl(w, 0, 32);

  const float4 v = ((const float4*)(H + (size_t)s * F))[lane];
  float* o = O + (size_t)d * F + lane * 4;
  gfadd(o + 0, w * v.x);
  gfadd(o + 1, w * v.y);
  gfadd(o + 2, w * v.z);
  gfadd(o + 3, w * v.w);
}

__global__ void gat_relu(float* __restrict__ O, int total)
{
  const int i = blockIdx.x * blockDim.x + threadIdx.x;
  if (i < total) O[i] = fmaxf(O[i], 0.f);
}

// ---------------------------------------------------------------------------
extern "C" void kernel_launch(void* const* d_in, const int* in_sizes, int n_in,
                              void* d_out, int out_size, void* d_ws, size_t ws_size,
                              hipStream_t stream)
{
  (void)n_in; (void)out_size; (void)ws_size;
  const float* x   = (const float*)d_in[0];
  const int*   ei  = (const int*)d_in[1];
  const float* W1  = (const float*)d_in[2];
  const float* a1s = (const float*)d_in[3];
  const float* a1d = (const float*)d_in[4];
  const float* b1  = (const float*)d_in[5];
  const float* W2  = (const float*)d_in[6];
  const float* a2s = (const float*)d_in[7];
  const float* a2d = (const float*)d_in[8];
  const float* b2  = (const float*)d_in[9];

  const int n = in_sizes[0] / F;     // 50000
  const int E = in_sizes[1] / 2;     // 1,600,000
  const int* src = ei;
  const int* dst = ei + E;
  const int Et = E + n;              // edges + self loops

  // workspace layout (floats): h[n*128] | o1[n*128] | asrc[n] | adst[n] | m[n] | sum[n]
  float*    h    = (float*)d_ws;
  float*    o1   = h + (size_t)n * F;
  float*    asrc = o1 + (size_t)n * F;
  float*    adst = asrc + n;
  unsigned* m    = (unsigned*)(adst + n);
  float*    sum  = (float*)(m + n);
  float*    out  = (float*)d_out;

  const dim3 gblk(32, 8, 1);
  const int gemm_grid = (n + 15) / 16;
  const int node_wave_grid = (int)(((size_t)n * 32 + 255) / 256);
  const int nf_grid   = (int)(((size_t)n * F + 255) / 256);
  const int edge_grid = (Et + 255) / 256;
  const int agg_grid  = (int)(((size_t)Et * 32 + 255) / 256);

  // ---- layer 1 ----
  gat_gemm_wmma<<<gemm_grid, gblk, 0, stream>>>(x, W1, h, n);
  gat_alphas<<<node_wave_grid, 256, 0, stream>>>(h, a1s, a1d, asrc, adst, n);
  gat_init<<<nf_grid, 256, 0, stream>>>(o1, b1, m, sum, n);
  gat_edge_max<<<edge_grid, 256, 0, stream>>>(src, dst, E, n, asrc, adst, m);
  gat_edge_sum<<<edge_grid, 256, 0, stream>>>(src, dst, E, n, asrc, adst, m, sum);
  gat_edge_agg<<<agg_grid, 256, 0, stream>>>(src, dst, E, n, asrc, adst, m, sum, h, o1);
  gat_relu<<<nf_grid, 256, 0, stream>>>(o1, n * F);

  // ---- layer 2 ----
  gat_gemm_wmma<<<gemm_grid, gblk, 0, stream>>>(o1, W2, h, n);
  gat_alphas<<<node_wave_grid, 256, 0, stream>>>(h, a2s, a2d, asrc, adst, n);
  gat_init<<<nf_grid, 256, 0, stream>>>(out, b2, m, sum, n);
  gat_edge_max<<<edge_grid, 256, 0, stream>>>(src, dst, E, n, asrc, adst, m);
  gat_edge_sum<<<edge_grid, 256, 0, stream>>>(src, dst, E, n, asrc, adst, m, sum);
  gat_edge_agg<<<agg_grid, 256, 0, stream>>>(src, dst, E, n, asrc, adst, m, sum, h, out);
}